// CrossDensityLoss_12807592477409
// MI455X (gfx1250) — compile-verified
//
#include <hip/hip_runtime.h>
#include <hip/hip_bf16.h>

typedef __attribute__((ext_vector_type(2))) float v2f;
typedef __attribute__((ext_vector_type(8))) float v8f;

#define KNN_WAVES 4
#define TEMP_INV 10.0f      // 1/0.1
#define FLT_BIG 3.4028235e38f

__device__ __forceinline__ int iminx(int a, int b) { return a < b ? a : b; }

// ---------------- per-row inverse L2 norm of 96-wide feature rows ----------
__global__ void rowinv_kernel(const float* __restrict__ feat, int N,
                              float* __restrict__ inv)
{
    int row  = blockIdx.x * (blockDim.x >> 5) + (threadIdx.x >> 5);
    int lane = threadIdx.x & 31;
    if (row >= N) return;
    const float* r = feat + (size_t)row * 96;
    float a = r[lane], b = r[lane + 32], c = r[lane + 64];
    float s = a * a + b * b + c * c;
#pragma unroll
    for (int off = 16; off > 0; off >>= 1) s += __shfl_xor(s, off, 32);
    if (lane == 0) inv[row] = 1.0f / fmaxf(sqrtf(s), 1e-12f);
}

// ---------------- squared norms of 3-d coords ------------------------------
__global__ void cc_kernel(const float* __restrict__ coord, int N,
                          float* __restrict__ cc)
{
    int i = blockIdx.x * blockDim.x + threadIdx.x;
    if (i >= N) return;
    float x = coord[3 * i], y = coord[3 * i + 1], z = coord[3 * i + 2];
    cc[i] = x * x + y * y + z * z;
}

// ---------------- KNN tile scan ------------------------------------------
// WMMA computes D = (-2*Q) x B^T + C, with C[m][n] = |b_n|^2 splat per lane
// column. dist = |q_m|^2 + D[m][n]. Tile transposed through LDS (20-float
// row stride: 16B-aligned rows -> b128 row reads, conflict-free stores).
template<bool CHECK>
__device__ __forceinline__ void knn_scan_range(
    int begin, int end, int Ndb,
    const float* __restrict__ coordDb, const float* __restrict__ ccDb,
    float* __restrict__ sRow,          // this wave's [32][20] tile
    int lane, int l, int half,
    v2f A0, v2f A1, float qq,
    float& d0, float& d1, float& d2, float& d3,
    int& i0, int& i1, int& i2, int& i3)
{
    for (int base = begin; base < end; base += 16) {
        int p = base + l;
        if (CHECK) p = iminx(p, Ndb - 1);
        float bx = coordDb[3 * p], by = coordDb[3 * p + 1], bz = coordDb[3 * p + 2];
        // B (4x16): lanes 0-15 hold rows K=0,1; lanes 16-31 rows K=2,3
        v2f B;
        B.x = half ? bz : bx;  B.y = half ? 0.0f : by;
        // C: lane owns column n = l for its 8 rows -> splat |b_n|^2
        float bb = ccDb[p];
        v8f C = {bb, bb, bb, bb, bb, bb, bb, bb};

        v8f Da = __builtin_amdgcn_wmma_f32_16x16x4_f32(false, A0, false, B,
                                                       (short)0, C, false, false);
        v8f Db = __builtin_amdgcn_wmma_f32_16x16x4_f32(false, A1, false, B,
                                                       (short)0, C, false, false);

        // D layout: lanes 0-15 -> rows 0..7 (vgpr r), col = lane; 16-31 -> rows 8..15
#pragma unroll
        for (int r = 0; r < 8; ++r) {
            sRow[(half * 8 + r) * 20 + l]      = Da[r];
            sRow[(16 + half * 8 + r) * 20 + l] = Db[r];
        }

        // per-wave LDS is in-order: b128 row read-back, then top-4 insert
        const float4* row4 = (const float4*)(sRow + lane * 20);
#pragma unroll
        for (int t4 = 0; t4 < 4; ++t4) {
            float4 v = row4[t4];
            float dv[4] = {v.x, v.y, v.z, v.w};
#pragma unroll
            for (int u = 0; u < 4; ++u) {
                int   dbi  = base + t4 * 4 + u;
                float dist = qq + dv[u];
                bool  ok   = CHECK ? (dbi < Ndb) : true;
                if (ok && dist < d3) {            // cheap execz skip
                    d3 = dist; i3 = dbi;          // replace current max
                    float tf; int ti;             // branchless bubble-up
                    bool c2 = d3 < d2;
                    tf = d2; ti = i2;
                    d2 = c2 ? d3 : d2; i2 = c2 ? i3 : i2;
                    d3 = c2 ? tf : d3; i3 = c2 ? ti : i3;
                    bool c1 = d2 < d1;
                    tf = d1; ti = i1;
                    d1 = c1 ? d2 : d1; i1 = c1 ? i2 : i1;
                    d2 = c1 ? tf : d2; i2 = c1 ? ti : i2;
                    bool c0 = d1 < d0;
                    tf = d0; ti = i0;
                    d0 = c0 ? d1 : d0; i0 = c0 ? i1 : i0;
                    d1 = c0 ? tf : d1; i1 = c0 ? ti : i1;
                }
            }
        }
    }
}

// ---------------- brute-force KNN (K=4) via f32 WMMA 16x16x4 ---------------
__global__ void knn_kernel(const float* __restrict__ coordQ,
                           const float* __restrict__ coordDb,
                           const float* __restrict__ ccQ,
                           const float* __restrict__ ccDb,
                           int Nq, int Ndb, int* __restrict__ knnOut)
{
    __shared__ __align__(16) float sD[KNN_WAVES][32][20];

    int wave = threadIdx.x >> 5;
    int lane = threadIdx.x & 31;
    int l    = lane & 15;
    int half = lane >> 4;
    int qBase = (blockIdx.x * KNN_WAVES + wave) * 32;

    // A (16x4 f32): lanes 0-15 hold K=0,1 (v0,v1); lanes 16-31 K=2,3.
    // Pre-scaled by -2 so the WMMA directly yields -2*q.b (+ C).
    int qa = iminx(qBase + l, Nq - 1);
    int qb = iminx(qBase + 16 + l, Nq - 1);
    float ax0 = coordQ[3 * qa], ay0 = coordQ[3 * qa + 1], az0 = coordQ[3 * qa + 2];
    float ax1 = coordQ[3 * qb], ay1 = coordQ[3 * qb + 1], az1 = coordQ[3 * qb + 2];
    v2f A0, A1;
    A0.x = -2.0f * (half ? az0 : ax0);  A0.y = half ? 0.0f : (-2.0f * ay0);
    A1.x = -2.0f * (half ? az1 : ax1);  A1.y = half ? 0.0f : (-2.0f * ay1);

    int   myq = qBase + lane;
    float qq  = ccQ[iminx(myq, Nq - 1)];

    float d0 = FLT_BIG, d1 = FLT_BIG, d2 = FLT_BIG, d3 = FLT_BIG;
    int   i0 = 0, i1 = 0, i2 = 0, i3 = 0;

    int nFull = Ndb & ~15;
    knn_scan_range<false>(0, nFull, Ndb, coordDb, ccDb, &sD[wave][0][0],
                          lane, l, half, A0, A1, qq,
                          d0, d1, d2, d3, i0, i1, i2, i3);
    if (nFull < Ndb)
        knn_scan_range<true>(nFull, Ndb, Ndb, coordDb, ccDb, &sD[wave][0][0],
                             lane, l, half, A0, A1, qq,
                             d0, d1, d2, d3, i0, i1, i2, i3);

    if (myq < Nq) {
        knnOut[4 * myq + 0] = i0;
        knnOut[4 * myq + 1] = i1;
        knnOut[4 * myq + 2] = i2;
        knnOut[4 * myq + 3] = i3;
    }
}

// ---------------- per-query softmax self-consistency loss ------------------
__global__ void loss_kernel(const float* __restrict__ featQ,
                            const float* __restrict__ featDb,
                            const float* __restrict__ invQ,
                            const float* __restrict__ invDb,
                            const int* __restrict__ knn, int Nq,
                            float* __restrict__ lossOut)
{
    int q    = blockIdx.x * (blockDim.x >> 5) + (threadIdx.x >> 5);
    int lane = threadIdx.x & 31;
    if (q >= Nq) return;

    const float* fq = featQ + (size_t)q * 96;
    float q0 = fq[lane], q1 = fq[lane + 32], q2 = fq[lane + 64];
    float iq = invQ[q];

    float sim[4];
#pragma unroll
    for (int k = 0; k < 4; ++k) {
        int idx = knn[q * 4 + k];
        const float* fd = featDb + (size_t)idx * 96;
        float a = q0 * fd[lane] + q1 * fd[lane + 32] + q2 * fd[lane + 64];
#pragma unroll
        for (int off = 16; off > 0; off >>= 1) a += __shfl_xor(a, off, 32);
        sim[k] = a * iq * invDb[idx] * TEMP_INV;
    }

    if (lane == 0) {
        float m  = fmaxf(fmaxf(sim[0], sim[1]), fmaxf(sim[2], sim[3]));
        float e0 = __expf(sim[0] - m), e1 = __expf(sim[1] - m);
        float e2 = __expf(sim[2] - m), e3 = __expf(sim[3] - m);
        float is = 1.0f / (e0 + e1 + e2 + e3);
        float p0 = e0 * is, p1 = e1 * is, p2 = e2 * is, p3 = e3 * is;
        float psq = p0 * p0 + p1 * p1 + p2 * p2 + p3 * p3;
        lossOut[q] = -__logf(psq + 1e-12f);
    }
}

// ---------------- deterministic final reduction ----------------------------
__global__ void reduce_kernel(const float* __restrict__ lossI, int Ni,
                              const float* __restrict__ lossJ, int Nj,
                              float* __restrict__ out)
{
    __shared__ float s[256];
    int t = threadIdx.x;

    float a = 0.0f;
    for (int i = t; i < Ni; i += 256) a += lossI[i];
    s[t] = a;
    __syncthreads();
    for (int st = 128; st > 0; st >>= 1) {
        if (t < st) s[t] += s[t + st];
        __syncthreads();
    }
    float sumI = (t == 0) ? s[0] : 0.0f;
    __syncthreads();

    float b = 0.0f;
    for (int i = t; i < Nj; i += 256) b += lossJ[i];
    s[t] = b;
    __syncthreads();
    for (int st = 128; st > 0; st >>= 1) {
        if (t < st) s[t] += s[t + st];
        __syncthreads();
    }
    if (t == 0)
        out[0] = 0.5f * (sumI / (float)Ni + s[0] / (float)Nj);
}

extern "C" void kernel_launch(void* const* d_in, const int* in_sizes, int n_in,
                              void* d_out, int out_size, void* d_ws, size_t ws_size,
                              hipStream_t stream)
{
    const float* feat_i  = (const float*)d_in[0];   // [Ni,96]
    const float* coord_i = (const float*)d_in[1];   // [Ni,3]
    const float* feat_j  = (const float*)d_in[2];   // [Nj,96]
    const float* coord_j = (const float*)d_in[3];   // [Nj,3]
    int Ni = in_sizes[1] / 3;
    int Nj = in_sizes[3] / 3;

    float* wsf = (float*)d_ws;
    size_t off = 0;
    float* inv_i  = wsf + off; off += (size_t)Ni;
    float* inv_j  = wsf + off; off += (size_t)Nj;
    float* cc_i   = wsf + off; off += (size_t)Ni;
    float* cc_j   = wsf + off; off += (size_t)Nj;
    int*   knn_i  = (int*)(wsf + off); off += (size_t)Ni * 4;
    int*   knn_j  = (int*)(wsf + off); off += (size_t)Nj * 4;
    float* loss_i = wsf + off; off += (size_t)Ni;
    float* loss_j = wsf + off; off += (size_t)Nj;

    rowinv_kernel<<<(Ni + 3) / 4, 128, 0, stream>>>(feat_i, Ni, inv_i);
    rowinv_kernel<<<(Nj + 3) / 4, 128, 0, stream>>>(feat_j, Nj, inv_j);
    cc_kernel<<<(Ni + 255) / 256, 256, 0, stream>>>(coord_i, Ni, cc_i);
    cc_kernel<<<(Nj + 255) / 256, 256, 0, stream>>>(coord_j, Nj, cc_j);

    int qpb = KNN_WAVES * 32;
    knn_kernel<<<(Ni + qpb - 1) / qpb, KNN_WAVES * 32, 0, stream>>>(
        coord_i, coord_j, cc_i, cc_j, Ni, Nj, knn_i);
    knn_kernel<<<(Nj + qpb - 1) / qpb, KNN_WAVES * 32, 0, stream>>>(
        coord_j, coord_i, cc_j, cc_i, Nj, Ni, knn_j);

    loss_kernel<<<(Ni + 3) / 4, 128, 0, stream>>>(feat_i, feat_j, inv_i, inv_j,
                                                  knn_i, Ni, loss_i);
    loss_kernel<<<(Nj + 3) / 4, 128, 0, stream>>>(feat_j, feat_i, inv_j, inv_i,
                                                  knn_j, Nj, loss_j);

    reduce_kernel<<<1, 256, 0, stream>>>(loss_i, Ni, loss_j, Nj, (float*)d_out);
}